// ArgDistanceMeasure_21131239096642
// MI455X (gfx1250) — compile-verified
//
#include <hip/hip_runtime.h>

// ---------------------------------------------------------------------------
// MI455X plan (bandwidth-bound):
//   FLOPs = 137 GFLOP f32;  HBM bytes ~= 132 MB (b stays L2-resident, 192MB L2)
//   -> ~5.7us memory floor vs <0.5us f32-WMMA compute -> keep exact f32 math,
//      use V_WMMA_F32_16X16X4_F32 for the cross term a@b.T.
// Squared-distance trick: sq = rt[m] + ct[j] - 2*cross, where
//   rt[m] = ||a_m||^2 + 2*eps*sum(a_m) + D*eps^2,  ct[j] = ||b_j||^2 - 2*eps*sum(b_j)
// sqrt is monotone -> top-k on sq == top-k on dist.
// ---------------------------------------------------------------------------

typedef __attribute__((ext_vector_type(2))) float v2f;
typedef __attribute__((ext_vector_type(8))) float v8f;

#define EPS_F      1e-6f
#define D_DIM      512
#define NA         2048
#define NB         65536
#define TOPN       16
#define NCHUNK     16
#define CHUNK      (NB / NCHUNK)        // 4096 b-cols per block
#define WAVES      8
#define COLS_ITER  (WAVES * 16)         // 128 b-cols per block iteration
#define ITERS      (CHUNK / COLS_ITER)  // 32
#define BBATCH     128                  // b_batch_size from setup_inputs
#define FLT_BIG    3.0e38f

// ---------------- phase 0: per-row correction terms (wave-per-row) ---------
template <bool IS_A>
__global__ void terms_kernel(const float* __restrict__ x, float* __restrict__ term) {
    const int wave = threadIdx.x >> 5;
    const int lane = threadIdx.x & 31;
    const int row  = blockIdx.x * 8 + wave;
    const float* p = x + (size_t)row * D_DIM;
    float s = 0.f, sq = 0.f;
    #pragma unroll 4
    for (int k = lane; k < D_DIM; k += 32) { float v = p[k]; s += v; sq += v * v; }
    #pragma unroll
    for (int off = 16; off > 0; off >>= 1) {
        s  += __shfl_xor(s,  off, 32);
        sq += __shfl_xor(sq, off, 32);
    }
    if (lane == 0) {
        term[row] = IS_A ? (sq + 2.0f * EPS_F * s + (float)D_DIM * EPS_F * EPS_F)
                         : (sq - 2.0f * EPS_F * s);
    }
}

// ---------------- phase 1: WMMA distance tiles + streaming per-row top-16 --
__global__ __launch_bounds__(256, 1)
void dist_topk_kernel(const float* __restrict__ a, const float* __restrict__ b,
                      const float* __restrict__ row_term,
                      const float* __restrict__ col_term,
                      float* __restrict__ cand_v, int* __restrict__ cand_i) {
    __shared__ float tile[16][COLS_ITER + 4];  // squared distances for this iter
    __shared__ float cvals[16][256];           // 16 partial top-16 lists per row
    __shared__ int   cidx[16][256];
    __shared__ float rterm_s[16];

    const int tid  = threadIdx.x;
    const int wave = tid >> 5;
    const int lane = tid & 31;
    const int lmod = lane & 15;
    const int lhi  = lane >> 4;                // 0: K0..1 / M0..7, 1: K2..3 / M8..15
    const int m0    = blockIdx.y * 16;
    const int chunk = blockIdx.x;
    const int chunk0 = chunk * CHUNK;

    if (tid < 16) rterm_s[tid] = row_term[m0 + tid];
    __syncthreads();

    // private streaming top-16 (sorted ascending), 16 threads per row
    float tv[TOPN]; int ti[TOPN];
    #pragma unroll
    for (int i = 0; i < TOPN; ++i) { tv[i] = FLT_BIG; ti[i] = 0; }
    const int selr = tid & 15;   // row this thread selects for
    const int selg = tid >> 4;   // scans cols [selg*8, selg*8+8)

    // A fragment base: lane n<16 holds a[m0+n][kk..kk+1]; lane n+16 -> kk+2..kk+3
    const float* arow = a + (size_t)(m0 + lmod) * D_DIM + lhi * 2;

    for (int it = 0; it < ITERS; ++it) {
        const int colbase = chunk0 + it * COLS_ITER;
        const int col0 = colbase + wave * 16;
        const float* brow = b + (size_t)(col0 + lmod) * D_DIM + lhi * 2;
        // prefetch next iteration's b rows into cache (global_prefetch_b8)
        if (it + 1 < ITERS)
            __builtin_prefetch(b + (size_t)(col0 + COLS_ITER + lmod) * D_DIM, 0, 1);

        v8f acc0 = {}; v8f acc1 = {};          // two chains for ILP
        #pragma unroll 4
        for (int kk = 0; kk < D_DIM; kk += 8) {
            v2f af0 = *(const v2f*)(arow + kk);
            v2f bf0 = *(const v2f*)(brow + kk);
            v2f af1 = *(const v2f*)(arow + kk + 4);
            v2f bf1 = *(const v2f*)(brow + kk + 4);
            acc0 = __builtin_amdgcn_wmma_f32_16x16x4_f32(
                false, af0, false, bf0, (short)0, acc0, false, false);
            acc1 = __builtin_amdgcn_wmma_f32_16x16x4_f32(
                false, af1, false, bf1, (short)0, acc1, false, false);
        }

        // C/D layout: VGPR v, lanes 0-15 -> M=v, lanes 16-31 -> M=v+8; N=lane%16
        const float ct = col_term[col0 + lmod];
        #pragma unroll
        for (int v = 0; v < 8; ++v) {
            const int M = v + lhi * 8;
            tile[M][wave * 16 + lmod] =
                rterm_s[M] + ct - 2.0f * (acc0[v] + acc1[v]);
        }
        __syncthreads();

        // selection: 16 threads per row, 8 cols each, early-out insertion
        #pragma unroll
        for (int u = 0; u < 8; ++u) {
            const int c = selg * 8 + u;
            float vv = tile[selr][c];
            if (vv < tv[TOPN - 1]) {
                int vi = colbase + c;
                #pragma unroll
                for (int i = 0; i < TOPN; ++i) {
                    if (vv < tv[i]) {
                        float tf = tv[i]; tv[i] = vv; vv = tf;
                        int   tt = ti[i]; ti[i] = vi; vi = tt;
                    }
                }
            }
        }
        __syncthreads();
    }

    // dump private lists, merge 16 lists -> chunk top-16 per row
    #pragma unroll
    for (int i = 0; i < TOPN; ++i) {
        cvals[selr][selg * 16 + i] = tv[i];
        cidx[selr][selg * 16 + i]  = ti[i];
    }
    __syncthreads();

    if (tid < 16) {
        float mv[TOPN]; int mi[TOPN];
        #pragma unroll
        for (int i = 0; i < TOPN; ++i) { mv[i] = FLT_BIG; mi[i] = 0; }
        for (int s = 0; s < 256; ++s) {
            float vv = cvals[tid][s];
            if (vv < mv[TOPN - 1]) {
                int vi = cidx[tid][s];
                #pragma unroll
                for (int i = 0; i < TOPN; ++i) {
                    if (vv < mv[i]) {
                        float tf = mv[i]; mv[i] = vv; vv = tf;
                        int   tt = mi[i]; mi[i] = vi; vi = tt;
                    }
                }
            }
        }
        const size_t base = ((size_t)(m0 + tid) * NCHUNK + chunk) * TOPN;
        for (int i = 0; i < TOPN; ++i) { cand_v[base + i] = mv[i]; cand_i[base + i] = mi[i]; }
    }
}

// ---------------- phase 2: merge chunks, apply buggy index remap -----------
__global__ void merge_kernel(const float* __restrict__ cand_v,
                             const int* __restrict__ cand_i,
                             long long* __restrict__ out) {
    const int row = blockIdx.x * blockDim.x + threadIdx.x;
    if (row >= NA) return;
    float mv[TOPN]; int mi[TOPN];
    #pragma unroll
    for (int i = 0; i < TOPN; ++i) { mv[i] = FLT_BIG; mi[i] = 0; }
    const float* cv = cand_v + (size_t)row * NCHUNK * TOPN;
    const int*   ci = cand_i + (size_t)row * NCHUNK * TOPN;
    for (int s = 0; s < NCHUNK * TOPN; ++s) {
        float vv = cv[s];
        if (vv < mv[TOPN - 1]) {
            int vi = ci[s];
            #pragma unroll
            for (int i = 0; i < TOPN; ++i) {
                if (vv < mv[i]) {
                    float tf = mv[i]; mv[i] = vv; vv = tf;
                    int   tt = mi[i]; mi[i] = vi; vi = tt;
                }
            }
        }
    }
    #pragma unroll
    for (int i = 0; i < TOPN; ++i) {
        const int idx = mi[i];
        out[(size_t)row * TOPN + i] =
            (long long)((idx % BBATCH) + (idx / BBATCH));
    }
}

// ---------------------------------------------------------------------------
extern "C" void kernel_launch(void* const* d_in, const int* in_sizes, int n_in,
                              void* d_out, int out_size, void* d_ws, size_t ws_size,
                              hipStream_t stream) {
    const float* a = (const float*)d_in[0];   // [2048, 512]
    const float* b = (const float*)d_in[1];   // [65536, 512]

    float* col_term = (float*)d_ws;                               // NB floats
    float* row_term = col_term + NB;                              // NA floats
    float* cand_v   = row_term + NA;                              // NA*NCHUNK*TOPN
    int*   cand_i   = (int*)(cand_v + (size_t)NA * NCHUNK * TOPN);
    long long* out  = (long long*)d_out;                          // [2048, 16] int64

    terms_kernel<false><<<NB / 8, 256, 0, stream>>>(b, col_term);
    terms_kernel<true ><<<NA / 8, 256, 0, stream>>>(a, row_term);

    dim3 grid(NCHUNK, NA / 16);   // 16 b-chunks x 128 a-row-blocks
    dist_topk_kernel<<<grid, 256, 0, stream>>>(a, b, row_term, col_term,
                                               cand_v, cand_i);

    merge_kernel<<<(NA + 255) / 256, 256, 0, stream>>>(cand_v, cand_i, out);
}